// HyperConnectionMatrix_85392539779780
// MI455X (gfx1250) — compile-verified
//
#include <hip/hip_runtime.h>

#define N 8192
#define NITER 10
#define SINK_EPS 1e-8f

typedef __attribute__((ext_vector_type(2))) float v2f;
typedef __attribute__((ext_vector_type(8))) float v8f;

// ---------------------------------------------------------------------------
// Kernel 1: E = exp(W).  64M elements, float4 vectorized, one streaming pass.
// ---------------------------------------------------------------------------
__global__ void sink_exp(const float* __restrict__ W, float* __restrict__ E) {
    size_t idx = (size_t)blockIdx.x * blockDim.x + threadIdx.x;  // float4 index
    float4 v = ((const float4*)W)[idx];
    float4 o;
    o.x = expf(v.x);
    o.y = expf(v.y);
    o.z = expf(v.z);
    o.w = expf(v.w);
    ((float4*)E)[idx] = o;
}

// ---------------------------------------------------------------------------
// Kernel 2: r[j] = c[j] = 1
// ---------------------------------------------------------------------------
__global__ void sink_init_rc(float* __restrict__ r, float* __restrict__ c) {
    int j = blockIdx.x * blockDim.x + threadIdx.x;
    r[j] = 1.0f;
    c[j] = 1.0f;
}

// ---------------------------------------------------------------------------
// Shared WMMA row-sum core for a 16-row strip (block = 256 threads, 8 waves).
// Wave w reduces columns [w*1024, (w+1)*1024) with V_WMMA_F32_16X16X4_F32:
//   A tile (16x4 fp32): lane m = lane&15 is row, kh = lane>>4 -> K pair,
//     each lane does one b64 load of E[i0+m][j+2kh .. +1].
//   B tile (4x16 fp32): B[k][n] = c[j+k] broadcast over N, so every column
//     of D carries the c-weighted row sums; extract column 0 (lanes 0/16).
// Leaves the 16 per-row sums in part[][], reduced by threads 0..15.
// ---------------------------------------------------------------------------
__device__ __forceinline__ void wmma_row_sums(const float* __restrict__ E,
                                              const float* __restrict__ c,
                                              int i0, float (*part)[16]) {
    const int tid  = threadIdx.x;
    const int lane = tid & 31;
    const int w    = tid >> 5;
    const int m    = lane & 15;
    const int kh   = lane >> 4;

    const float* rowp = E + (size_t)(i0 + m) * N + 2 * kh;
    const float* cp   = c + 2 * kh;

    v8f acc = {};
    const int jbeg = w * 1024;
    const int jend = jbeg + 1024;
    for (int j = jbeg; j < jend; j += 4) {
        if ((j & 255) == 0) {
            __builtin_prefetch(rowp + j + 2048, 0, 1);  // global_prefetch_b8
        }
        v2f a = *(const v2f*)(rowp + j);
        v2f b = *(const v2f*)(cp + j);
        acc = __builtin_amdgcn_wmma_f32_16x16x4_f32(
            /*neg_a=*/false, a, /*neg_b=*/false, b,
            /*c_mod=*/(short)0, acc, /*reuse_a=*/false, /*reuse_b=*/false);
    }
    if (lane == 0) {
        #pragma unroll
        for (int v = 0; v < 8; ++v) part[w][v] = acc[v];
    } else if (lane == 16) {
        #pragma unroll
        for (int v = 0; v < 8; ++v) part[w][8 + v] = acc[v];
    }
}

// ---------------------------------------------------------------------------
// FUSED iteration kernel (preferred path): block owns a 16-row strip.
// Pass 1: WMMA row sums -> update r for the strip's rows (strip read once).
// Pass 2: column partials of the SAME strip with the freshly updated r
//         (strip re-read while still cache-hot), written deterministically
//         to tpart[block][j].  One effective HBM pass per iteration.
// ---------------------------------------------------------------------------
__global__ void sink_fused_iter(const float* __restrict__ E,
                                const float* __restrict__ c,
                                float* __restrict__ r,
                                float* __restrict__ tpart) {
    __shared__ float part[8][16];
    __shared__ float newr[16];

    const int tid = threadIdx.x;
    const int i0  = blockIdx.x * 16;

    wmma_row_sums(E, c, i0, part);
    __syncthreads();

    if (tid < 16) {
        float s = 0.0f;
        #pragma unroll
        for (int ww = 0; ww < 8; ++ww) s += part[ww][tid];
        float rv = r[i0 + tid];
        rv = rv / (rv * s + SINK_EPS);
        r[i0 + tid] = rv;
        newr[tid] = rv;
    }
    __syncthreads();

    float rr[16];
    #pragma unroll
    for (int ii = 0; ii < 16; ++ii) rr[ii] = newr[ii];

    const float* base = E + (size_t)i0 * N;
    float* tp = tpart + (size_t)blockIdx.x * N;
    for (int k = 0; k < 32; ++k) {           // 32 coalesced column groups
        const int j = k * 256 + tid;
        float a0 = 0.f, a1 = 0.f, a2 = 0.f, a3 = 0.f;
        #pragma unroll
        for (int ii = 0; ii < 16; ii += 4) {
            a0 += base[(size_t)(ii + 0) * N + j] * rr[ii + 0];
            a1 += base[(size_t)(ii + 1) * N + j] * rr[ii + 1];
            a2 += base[(size_t)(ii + 2) * N + j] * rr[ii + 2];
            a3 += base[(size_t)(ii + 3) * N + j] * rr[ii + 3];
        }
        tp[j] = (a0 + a1) + (a2 + a3);
    }
}

// ---------------------------------------------------------------------------
// Fallback path kernels (small workspace): row update only, then col partials
// over 32 fat row strips (identical math, two passes over E per iteration).
// ---------------------------------------------------------------------------
__global__ void sink_row_update(const float* __restrict__ E,
                                const float* __restrict__ c,
                                float* __restrict__ r) {
    __shared__ float part[8][16];
    const int tid = threadIdx.x;
    const int i0  = blockIdx.x * 16;

    wmma_row_sums(E, c, i0, part);
    __syncthreads();

    if (tid < 16) {
        float s = 0.0f;
        #pragma unroll
        for (int ww = 0; ww < 8; ++ww) s += part[ww][tid];
        float rv = r[i0 + tid];
        r[i0 + tid] = rv / (rv * s + SINK_EPS);
    }
}

__global__ void sink_col_partial(const float* __restrict__ E,
                                 const float* __restrict__ r,
                                 float* __restrict__ tpart) {
    const int j     = blockIdx.x * 256 + threadIdx.x;
    const int rs    = blockIdx.y;
    const int ibase = rs * 256;
    const float* p  = E + (size_t)ibase * N + j;

    float a0 = 0.f, a1 = 0.f, a2 = 0.f, a3 = 0.f;
    for (int ii = 0; ii < 256; ii += 4) {
        a0 += p[(size_t)(ii + 0) * N] * r[ibase + ii + 0];
        a1 += p[(size_t)(ii + 1) * N] * r[ibase + ii + 1];
        a2 += p[(size_t)(ii + 2) * N] * r[ibase + ii + 2];
        a3 += p[(size_t)(ii + 3) * N] * r[ibase + ii + 3];
    }
    tpart[(size_t)rs * N + j] = (a0 + a1) + (a2 + a3);
}

// ---------------------------------------------------------------------------
// Finish column step: t_j = sum over nstrips partials; c <- c/(c*t + eps).
// nstrips = 512 (fused) or 32 (fallback); fixed order -> deterministic.
// ---------------------------------------------------------------------------
__global__ void sink_col_finish(const float* __restrict__ tpart,
                                float* __restrict__ c, int nstrips) {
    const int j = blockIdx.x * 256 + threadIdx.x;
    float t0 = 0.f, t1 = 0.f, t2 = 0.f, t3 = 0.f;
    for (int b = 0; b < nstrips; b += 4) {
        t0 += tpart[(size_t)(b + 0) * N + j];
        t1 += tpart[(size_t)(b + 1) * N + j];
        t2 += tpart[(size_t)(b + 2) * N + j];
        t3 += tpart[(size_t)(b + 3) * N + j];
    }
    float t  = (t0 + t1) + (t2 + t3);
    float cv = c[j];
    c[j] = cv / (cv * t + SINK_EPS);
}

// ---------------------------------------------------------------------------
// Kernel: P[i][j] = r[i] * E[i][j] * c[j], in place on d_out.
// ---------------------------------------------------------------------------
__global__ void sink_finalize(float* __restrict__ E,
                              const float* __restrict__ r,
                              const float* __restrict__ c) {
    size_t idx = (size_t)blockIdx.x * blockDim.x + threadIdx.x;  // float4 index
    const int i  = (int)(idx >> 11);   // 2048 float4 per row
    const int j4 = (int)(idx & 2047);
    float4 v  = ((float4*)E)[idx];
    float  rv = r[i];
    float4 cj = ((const float4*)c)[j4];
    v.x *= rv * cj.x;
    v.y *= rv * cj.y;
    v.z *= rv * cj.z;
    v.w *= rv * cj.w;
    ((float4*)E)[idx] = v;
}

// ---------------------------------------------------------------------------
extern "C" void kernel_launch(void* const* d_in, const int* in_sizes, int n_in,
                              void* d_out, int out_size, void* d_ws, size_t ws_size,
                              hipStream_t stream) {
    (void)in_sizes; (void)n_in; (void)out_size;

    const float* W = (const float*)d_in[0];
    float* E = (float*)d_out;          // 8192*8192 fp32 — doubles as exp(W) store

    float* r     = (float*)d_ws;       // 8192 floats
    float* cvec  = r + N;              // 8192 floats
    float* tpart = cvec + N;           // partial col sums

    // Fused path needs 512 partial strips (16.1 MB); fallback needs 32 (1.1 MB).
    const size_t need_fused = ((size_t)2 * N + (size_t)512 * N) * sizeof(float);
    const bool   use_fused  = (ws_size >= need_fused);

    const size_t elems4 = (size_t)N * N / 4;

    sink_exp<<<(unsigned)(elems4 / 256), 256, 0, stream>>>(W, E);
    sink_init_rc<<<N / 256, 256, 0, stream>>>(r, cvec);

    if (use_fused) {
        for (int it = 0; it < NITER; ++it) {
            sink_fused_iter<<<N / 16, 256, 0, stream>>>(E, cvec, r, tpart);
            sink_col_finish<<<N / 256, 256, 0, stream>>>(tpart, cvec, 512);
        }
    } else {
        for (int it = 0; it < NITER; ++it) {
            sink_row_update<<<N / 16, 256, 0, stream>>>(E, cvec, r);
            sink_col_partial<<<dim3(N / 256, 32), 256, 0, stream>>>(E, r, tpart);
            sink_col_finish<<<N / 256, 256, 0, stream>>>(tpart, cvec, 32);
        }
    }

    sink_finalize<<<(unsigned)(elems4 / 256), 256, 0, stream>>>(E, r, cvec);
}